// ScaledDotProductAttention_42460046688741
// MI455X (gfx1250) — compile-verified
//
#include <hip/hip_runtime.h>
#include <hip/hip_bf16.h>
#include <math.h>

// ---------------------------------------------------------------------------
// Fused scaled-dot-product attention for gfx1250 (MI455X), fp32 end-to-end.
//   scores = (Q K^T) * rsqrt(dk)  -> softmax -> *mask -> renormalize -> attn
//   context = attn @ V
// Outputs: d_out = [context (B,H,S,DK)] ++ [attn (B,H,S,S)]
// HBM-write bound (~1.07 GB attn), so fp32 WMMA (v_wmma_f32_16x16x4_f32)
// is both exact and free.
// ---------------------------------------------------------------------------

typedef __attribute__((ext_vector_type(2))) float v2f;
typedef __attribute__((ext_vector_type(8))) float v8f;

constexpr int SEQ   = 2048;
constexpr int DKDIM = 64;
constexpr int NBH   = 64;     // B*H = 4*16
constexpr int ROWS  = 16;     // query rows per block (one WMMA M-tile)
constexpr int WAVES = 8;      // 256 threads / wave32
constexpr int JTILES = (SEQ / 16) / WAVES;  // 16 column tiles per wave

__device__ __forceinline__ v8f wmma_f32(v2f a, v2f b, v8f c) {
  // D(16x16,f32) = A(16x4,f32) * B(4x16,f32) + C
  return __builtin_amdgcn_wmma_f32_16x16x4_f32(
      /*neg_a=*/false, a, /*neg_b=*/false, b,
      /*c_mod=*/(short)0, c, /*reuse_a=*/false, /*reuse_b=*/false);
}

__global__ __launch_bounds__(256)
void sdpa_fused_kernel(const float* __restrict__ Q,
                       const float* __restrict__ K,
                       const float* __restrict__ V,
                       const float* __restrict__ MASK,  // (S,S), broadcast over b,h
                       const int*   __restrict__ dkp,
                       float* __restrict__ ctx_out,     // (B,H,S,DK)
                       float* __restrict__ attn_out)    // (B,H,S,S)
{
  // --- LDS (≈58 KB total, bank-conflict-free padded strides) ---
  __shared__ float red [WAVES][32][8];        // per-lane partials (max / sum_all)
  __shared__ float red2[WAVES][32][8];        // per-lane partials (sum_masked)
  __shared__ float tr  [WAVES][16][17];       // per-wave A-tile transpose buffer
  __shared__ float cred[WAVES][4][8][32];     // split-K context partials
  __shared__ float rowmax[ROWS];
  __shared__ float rowinv[ROWS];

  const int tid  = threadIdx.x;
  const int w    = tid >> 5;      // wave id 0..7
  const int lane = tid & 31;
  const int half = lane >> 4;     // 0: lanes 0-15, 1: lanes 16-31
  const int nl   = lane & 15;

  const int bh = blockIdx.y;
  const int r0 = blockIdx.x * ROWS;

  const float scale = rsqrtf((float)dkp[0]);

  const float* q    = Q + ((size_t)bh * SEQ + r0) * DKDIM;
  const float* kmat = K + (size_t)bh * SEQ * DKDIM;
  const float* vmat = V + (size_t)bh * SEQ * DKDIM;
  const float* mrow = MASK + (size_t)r0 * SEQ;
  float* ao = attn_out + ((size_t)bh * SEQ + r0) * SEQ;
  float* co = ctx_out  + ((size_t)bh * SEQ + r0) * DKDIM;

  // ---- Q fragments (A-matrix 16x4 fp32 layout), scale folded in.
  // A layout: lane L holds row M=L&15; VGPR0 = K + 2*(L>>4), VGPR1 = +1.
  v2f qa[16];
#pragma unroll
  for (int kk = 0; kk < 16; ++kk) {
    const float* p = q + nl * DKDIM + 4 * kk + 2 * half;
    v2f t; t.x = p[0] * scale; t.y = p[1] * scale;
    qa[kk] = t;
  }

  // ---- Phase A: scores = Q K^T * scale for this wave's 16 column tiles.
  v8f  sarr[JTILES];      // register-resident 16x16 tiles (D-layout)
  float pmax[8];
#pragma unroll
  for (int i = 0; i < 8; ++i) pmax[i] = -3.0e38f;

#pragma unroll
  for (int j = 0; j < JTILES; ++j) {
    const int cbase = 16 * (w + WAVES * j);
    v8f acc = {};
#pragma unroll
    for (int kk = 0; kk < 16; ++kk) {
      // B-matrix 4x16 layout mirrors A: lane holds column N=L&15,
      // VGPR0 = K-row (kk*4 + 2*half), VGPR1 = +1.  B = K^T -> read K rows.
      const float* p = kmat + (size_t)(cbase + nl) * DKDIM + 4 * kk + 2 * half;
      v2f b; b.x = p[0]; b.y = p[1];
      acc = wmma_f32(qa[kk], b, acc);
    }
    sarr[j] = acc;
#pragma unroll
    for (int i = 0; i < 8; ++i) pmax[i] = fmaxf(pmax[i], acc[i]);
  }

  // ---- Row-max reduction (D-layout: vgpr i + 8*half = local row).
#pragma unroll
  for (int i = 0; i < 8; ++i) red[w][lane][i] = pmax[i];
  __syncthreads();
  if (tid < ROWS) {
    const int i  = tid & 7;
    const int l0 = (tid >> 3) << 4;
    float m = -3.0e38f;
    for (int w2 = 0; w2 < WAVES; ++w2)
      for (int l2 = 0; l2 < 16; ++l2)
        m = fmaxf(m, red[w2][l0 + l2][i]);
    rowmax[tid] = m;
  }
  __syncthreads();

  // ---- Phase B: e = exp(s - max); Z_all = Σe; Z_mask = Σ(e*mask); keep e*mask.
  float psall[8], psmask[8];
#pragma unroll
  for (int i = 0; i < 8; ++i) { psall[i] = 0.f; psmask[i] = 0.f; }
#pragma unroll
  for (int j = 0; j < JTILES; ++j) {
    const int col = 16 * (w + WAVES * j) + nl;
#pragma unroll
    for (int i = 0; i < 8; ++i) {
      const int row = i + 8 * half;
      float e = __expf(sarr[j][i] - rowmax[row]);
      psall[i] += e;
      float mv = mrow[(size_t)row * SEQ + col];
      float em = e * mv;
      psmask[i] += em;
      sarr[j][i] = em;
    }
  }
#pragma unroll
  for (int i = 0; i < 8; ++i) { red[w][lane][i] = psall[i]; red2[w][lane][i] = psmask[i]; }
  __syncthreads();
  if (tid < ROWS) {
    const int i  = tid & 7;
    const int l0 = (tid >> 3) << 4;
    float za = 0.f, zm = 0.f;
    for (int w2 = 0; w2 < WAVES; ++w2)
      for (int l2 = 0; l2 < 16; ++l2) { za += red[w2][l0 + l2][i]; zm += red2[w2][l0 + l2][i]; }
    // attn = (e/Z_all)*mask / (Z_mask/Z_all + 1e-8)  ==  e*mask / (Z_mask + 1e-8*Z_all)
    rowinv[tid] = 1.0f / (zm + 1e-8f * za);
  }
  __syncthreads();

  // ---- Phase C/D: normalize -> write attn to HBM -> ctx += A @ V (split-K).
  v8f cacc[4];
#pragma unroll
  for (int n = 0; n < 4; ++n) { v8f z = {}; cacc[n] = z; }

#pragma unroll
  for (int j = 0; j < JTILES; ++j) {
    const int cbase = 16 * (w + WAVES * j);
    const int col   = cbase + nl;
#pragma unroll
    for (int i = 0; i < 8; ++i) {
      const int row = i + 8 * half;
      float a = sarr[j][i] * rowinv[row];
      ao[(size_t)row * SEQ + col] = a;        // final attn element
      tr[w][row][nl] = a;                     // stage for A-fragment re-layout
    }
    // Wave-private LDS RAW; compiler inserts s_wait_dscnt.
#pragma unroll
    for (int kk = 0; kk < 4; ++kk) {
      v2f af;
      af.x = tr[w][nl][4 * kk + 2 * half];
      af.y = tr[w][nl][4 * kk + 2 * half + 1];
      const int krow = cbase + 4 * kk + 2 * half;
#pragma unroll
      for (int n = 0; n < 4; ++n) {
        v2f bf;
        bf.x = vmat[(size_t)krow * DKDIM + 16 * n + nl];
        bf.y = vmat[(size_t)(krow + 1) * DKDIM + 16 * n + nl];
        cacc[n] = wmma_f32(af, bf, cacc[n]);
      }
    }
  }

  // ---- Phase E: cross-wave split-K reduction of context, write out.
#pragma unroll
  for (int n = 0; n < 4; ++n)
#pragma unroll
    for (int i = 0; i < 8; ++i)
      cred[w][n][i][lane] = cacc[n][i];
  __syncthreads();

  for (int e2 = 0; e2 < 4; ++e2) {
    const int idx = tid + 256 * e2;           // 16*64 = 1024 elements
    const int r = idx >> 6, c = idx & 63;
    const int n = c >> 4, nn = c & 15;
    const int i = r & 7;
    const int ln = ((r >> 3) << 4) + nn;
    float s = 0.f;
    for (int w2 = 0; w2 < WAVES; ++w2) s += cred[w2][n][i][ln];
    co[(size_t)r * DKDIM + c] = s;
  }
}

extern "C" void kernel_launch(void* const* d_in, const int* in_sizes, int n_in,
                              void* d_out, int out_size, void* d_ws, size_t ws_size,
                              hipStream_t stream) {
  const float* Q    = (const float*)d_in[0];
  const float* K    = (const float*)d_in[1];
  const float* V    = (const float*)d_in[2];
  const float* MASK = (const float*)d_in[3];
  const int*   dk   = (const int*)d_in[4];

  float* out  = (float*)d_out;
  float* ctx  = out;                                            // (B,H,S,DK) first
  float* attn = out + (size_t)NBH * SEQ * DKDIM;                // then (B,H,S,S)

  dim3 grid(SEQ / ROWS, NBH);
  sdpa_fused_kernel<<<grid, 256, 0, stream>>>(Q, K, V, MASK, dk, ctx, attn);
}